// QuantumHybridBinaryClassifier_65481071401220
// MI455X (gfx1250) — compile-verified
//
#include <hip/hip_runtime.h>
#include <math.h>

// ---------------------------------------------------------------------------
// QuantumHybridBinaryClassifier — dead-code-eliminated CDNA5 kernel.
//
// The reference's conv/pool/fc1/fc2 pipeline is dead: the output depends ONLY
// on q_params, q_basis, fc3_w, fc3_b. Every output row is the same
// (p, 1-p) pair where p = sigmoid(evs . fc3_w + fc3_b) and evs comes from a
// 4-qubit (16 complex amplitude) state-vector simulation.
//
// Each 32-lane wave: (1) simulates the circuit fully unrolled in registers,
// (2) computes the fc3 reduction with V_WMMA_F32_16X16X4_F32 (A row m holds
// the 4 products ev[k]*w[k], B = all-ones, so D = sum of products in every
// element -- layout-permutation-proof), (3) stores 32 rows of (p, 1-p).
// ---------------------------------------------------------------------------

typedef __attribute__((ext_vector_type(2))) float v2f;
typedef __attribute__((ext_vector_type(8))) float v8f;

#define NQ 4
#define DEPTH 2

// axis w of the (2,2,2,2) state tensor <-> bit (3-w) of the flat index.

template <int BIT>
__device__ __forceinline__ void gate_rx(float* sr, float* si, float t) {
  float c = __cosf(0.5f * t), s = __sinf(0.5f * t);
#pragma unroll
  for (int i = 0; i < 16; ++i) {
    if ((i & (1 << BIT)) == 0) {
      const int j = i | (1 << BIT);
      float ari = sr[i], aii = si[i], arj = sr[j], aij = si[j];
      // [[c, -i s], [-i s, c]]
      sr[i] = fmaf(c, ari,  s * aij);
      si[i] = fmaf(c, aii, -s * arj);
      sr[j] = fmaf(c, arj,  s * aii);
      si[j] = fmaf(c, aij, -s * ari);
    }
  }
}

template <int BIT>
__device__ __forceinline__ void gate_ry(float* sr, float* si, float t) {
  float c = __cosf(0.5f * t), s = __sinf(0.5f * t);
#pragma unroll
  for (int i = 0; i < 16; ++i) {
    if ((i & (1 << BIT)) == 0) {
      const int j = i | (1 << BIT);
      float ari = sr[i], aii = si[i], arj = sr[j], aij = si[j];
      // [[c, -s], [s, c]] (real)
      sr[i] = fmaf(c, ari, -s * arj);
      si[i] = fmaf(c, aii, -s * aij);
      sr[j] = fmaf(s, ari,  c * arj);
      si[j] = fmaf(s, aii,  c * aij);
    }
  }
}

template <int BIT>
__device__ __forceinline__ void gate_rz(float* sr, float* si, float t) {
  float c = __cosf(0.5f * t), s = __sinf(0.5f * t);
#pragma unroll
  for (int i = 0; i < 16; ++i) {
    if ((i & (1 << BIT)) == 0) {
      const int j = i | (1 << BIT);
      float ari = sr[i], aii = si[i], arj = sr[j], aij = si[j];
      // diag(c - i s, c + i s)
      sr[i] = fmaf(c, ari,  s * aii);
      si[i] = fmaf(c, aii, -s * ari);
      sr[j] = fmaf(c, arj, -s * aij);
      si[j] = fmaf(c, aij,  s * arj);
    }
  }
}

template <int W>  // CNOT control=qubit W (bit 3-W), target=qubit W+1 (bit 2-W)
__device__ __forceinline__ void gate_cnot(float* sr, float* si) {
  constexpr int BC = 3 - W, BT = 2 - W;
#pragma unroll
  for (int i = 0; i < 16; ++i) {
    if (((i >> BC) & 1) == 1 && ((i >> BT) & 1) == 0) {
      const int j = i | (1 << BT);
      float tr = sr[i]; sr[i] = sr[j]; sr[j] = tr;
      float ti = si[i]; si[i] = si[j]; si[j] = ti;
    }
  }
}

__device__ __forceinline__ void rot_layer(float* sr, float* si, const float* p) {
  // p = 12 angles laid out [w][3] ; qubit w acts on bit 3-w
  gate_rx<3>(sr, si, p[0]);  gate_ry<3>(sr, si, p[1]);  gate_rz<3>(sr, si, p[2]);
  gate_rx<2>(sr, si, p[3]);  gate_ry<2>(sr, si, p[4]);  gate_rz<2>(sr, si, p[5]);
  gate_rx<1>(sr, si, p[6]);  gate_ry<1>(sr, si, p[7]);  gate_rz<1>(sr, si, p[8]);
  gate_rx<0>(sr, si, p[9]);  gate_ry<0>(sr, si, p[10]); gate_rz<0>(sr, si, p[11]);
}

__global__ void __launch_bounds__(32)
qhbc_kernel(const float* __restrict__ q_params,  // [DEPTH,NQ,3] = 24
            const float* __restrict__ q_basis,   // [NQ,3]       = 12
            const float* __restrict__ fc3_w,     // [1,NQ]       = 4
            const float* __restrict__ fc3_b,     // [1]
            float* __restrict__ out,             // [B,2]
            int B) {
  // ---- 16-amplitude state-vector simulation (uniform across all lanes) ----
  float sr[16], si[16];
#pragma unroll
  for (int i = 0; i < 16; ++i) { sr[i] = 0.0f; si[i] = 0.0f; }
  sr[0] = 1.0f;

  rot_layer(sr, si, q_basis);          // learnable basis rotation
#pragma unroll
  for (int layer = 0; layer < DEPTH; ++layer) {
    rot_layer(sr, si, q_params + layer * 12);
    gate_cnot<0>(sr, si);
    gate_cnot<1>(sr, si);
    gate_cnot<2>(sr, si);
  }

  // ---- Z expectation values per qubit ----
  float ev0 = 0.f, ev1 = 0.f, ev2 = 0.f, ev3 = 0.f;
#pragma unroll
  for (int i = 0; i < 16; ++i) {
    float p = fmaf(sr[i], sr[i], si[i] * si[i]);
    ev0 += ((i >> 3) & 1) ? -p : p;
    ev1 += ((i >> 2) & 1) ? -p : p;
    ev2 += ((i >> 1) & 1) ? -p : p;
    ev3 += ((i >> 0) & 1) ? -p : p;
  }

  // ---- fc3 reduction via V_WMMA_F32_16X16X4_F32 ----
  // A (16x4): row m holds the 4 products ev[k]*w[k]; documented f32 A layout
  // puts K={0,1} in lanes 0-15 and K={2,3} in lanes 16-31 (VGPR0/VGPR1).
  // B (4x16) = all ones  =>  D[m,n] = sum_k ev[k]*w[k]  in every element,
  // independent of any k-permutation in the operand layouts.
  float p0 = ev0 * fc3_w[0];
  float p1 = ev1 * fc3_w[1];
  float p2 = ev2 * fc3_w[2];
  float p3 = ev3 * fc3_w[3];

  const int lane = threadIdx.x;            // blockDim.x == 32, one full wave
  const bool lo = lane < 16;
  v2f a; a.x = lo ? p0 : p2; a.y = lo ? p1 : p3;
  v2f b; b.x = 1.0f; b.y = 1.0f;
  v8f c = {};
  // 8 args: (neg_a, A, neg_b, B, c_mod, C, reuse_a, reuse_b)
  v8f d = __builtin_amdgcn_wmma_f32_16x16x4_f32(
      false, a, false, b, (short)0, c, false, false);

  float logit = d[0] + fc3_b[0];
  float prob  = 1.0f / (1.0f + __expf(-logit));

  // ---- broadcast (p, 1-p) to this block's 32 batch rows ----
  const int row = blockIdx.x * 32 + lane;
  if (row < B) {
    float2 o = make_float2(prob, 1.0f - prob);
    *reinterpret_cast<float2*>(out + 2 * row) = o;
  }
}

extern "C" void kernel_launch(void* const* d_in, const int* in_sizes, int n_in,
                              void* d_out, int out_size, void* d_ws, size_t ws_size,
                              hipStream_t stream) {
  // setup_inputs order:
  // 0:x 1:conv1_w 2:conv1_b 3:conv2_w 4:conv2_b 5:fc1_w 6:fc1_b 7:fc2_w
  // 8:fc2_b 9:q_params 10:q_basis 11:fc3_w 12:fc3_b
  const float* q_params = (const float*)d_in[9];
  const float* q_basis  = (const float*)d_in[10];
  const float* fc3_w    = (const float*)d_in[11];
  const float* fc3_b    = (const float*)d_in[12];
  float* out = (float*)d_out;

  const int B = out_size / 2;              // [B,2] output
  const int grid = (B + 31) / 32;          // one wave per 32 rows
  qhbc_kernel<<<grid, 32, 0, stream>>>(q_params, q_basis, fc3_w, fc3_b, out, B);
}